// Capsule_2731599200537
// MI455X (gfx1250) — compile-verified
//
#include <hip/hip_runtime.h>

typedef float v2f __attribute__((ext_vector_type(2)));
typedef float v8f __attribute__((ext_vector_type(8)));

#define B_DIM     64
#define N_IN_     2048
#define IN_ATOMS_ 8
#define N_OUT_    32
#define OUT_A_    16
#define N_O       (N_OUT_ * OUT_A_)       // 512
#define NROUTE    3
#define I_PER_WG  16
#define NPART     (N_IN_ / I_PER_WG)      // 128 partial-sum slots
#define PRE_N     (B_DIM * N_OUT_ * OUT_A_)  // 32768
#define VROW      516                     // padded LDS row stride (floats)

// ---- CDNA5 async global->LDS copy (ASYNCcnt-tracked, no VGPR staging) -------
__device__ __forceinline__ void async_lds_b128(unsigned lds_byte_off,
                                               const void* gaddr) {
  asm volatile("global_load_async_to_lds_b128 %0, %1, off"
               :: "v"(lds_byte_off), "v"(gaddr) : "memory");
}
__device__ __forceinline__ void wait_asynccnt0() {
  asm volatile("s_wait_asynccnt 0x0" ::: "memory");
}

// ---------------------------------------------------------------- zero logits
__global__ __launch_bounds__(256) void k_init(float* __restrict__ p, int n) {
  int idx = blockIdx.x * blockDim.x + threadIdx.x;
  int stride = gridDim.x * blockDim.x;
  for (int i = idx; i < n; i += stride) p[i] = 0.0f;
}

// ------------------------------------------------- votes(WMMA) + route kernel
// One block = 16 consecutive input capsules i. Per i:
//   W[i] (16KB) -> LDS via global_load_async_to_lds_b128
//   votes[64x512] = x[:,i,:] (64x8) @ W[i] (8x512) via V_WMMA_F32_16X16X4_F32 -> LDS
//   logits += <votes, act_prev> (r>0), leaky softmax over j,
//   preact partials accumulated in registers, stored to this block's slot.
__global__ __launch_bounds__(256) void k_route(
    const float* __restrict__ x, const float* __restrict__ W,
    float* __restrict__ logits, const float* __restrict__ act_in,
    float* __restrict__ partial, int r)
{
  __shared__ float votes[B_DIM * VROW];        // 132 KB
  __shared__ float wlds[IN_ATOMS_ * N_O];      //  16 KB  (W[i] staged)

  const int tid  = threadIdx.x;
  const int lane = tid & 31;
  const int wave = tid >> 5;                   // 8 wave32s
  const int mt      = wave >> 1;               // M tile (batch rows)
  const int nt_base = (wave & 1) * 16;         // 16 N tiles per wave
  const int i_base  = blockIdx.x * I_PER_WG;

  const int bb = tid >> 2;                     // batch row for routing phase
  const int jq = tid & 3;                      // capsule quarter

  const unsigned wlds_base = (unsigned)(uintptr_t)(&wlds[0]);  // LDS byte offset

  float acc[8][16];
  #pragma unroll
  for (int jj = 0; jj < 8; ++jj)
    #pragma unroll
    for (int a = 0; a < 16; ++a) acc[jj][a] = 0.0f;

  for (int ii = 0; ii < I_PER_WG; ++ii) {
    const int i = i_base + ii;
    const float* Wi = W + (size_t)i * IN_ATOMS_ * N_O;

    // ---- stage W[i] (16 KB) into LDS: 1024 x B128 async copies
    {
      const char* gsrc = (const char*)Wi;
      #pragma unroll
      for (int p = 0; p < 4; ++p) {
        const unsigned off = (unsigned)(tid + p * 256) * 16u;
        async_lds_b128(wlds_base + off, gsrc + off);
      }
      wait_asynccnt0();
    }
    __syncthreads();   // W[i] visible to all waves in LDS

    // ---- A fragments: 16x4 f32 layout (ISA 7.12.2)
    // lanes 0-15: M=lane, K={0,1}; lanes 16-31: M=lane-16, K={2,3}
    const int am = mt * 16 + (lane & 15);
    const int ak = 2 * (lane >> 4);
    const float* xrow = x + ((size_t)am * N_IN_ + i) * IN_ATOMS_;
    v2f a0, a1;
    a0[0] = xrow[ak + 0];     a0[1] = xrow[ak + 1];       // K chunk 0..3
    a1[0] = xrow[4 + ak + 0]; a1[1] = xrow[4 + ak + 1];   // K chunk 4..7

    const int bn = lane & 15;
    const int bk = 2 * (lane >> 4);   // B: VGPR v holds row K = v + 2*(lane>=16)

    #pragma unroll
    for (int t = 0; t < 16; ++t) {
      const int nt = nt_base + t;
      const int n  = nt * 16 + bn;
      v2f b0, b1;
      b0[0] = wlds[(bk + 0) * N_O + n];
      b0[1] = wlds[(bk + 1) * N_O + n];
      b1[0] = wlds[(4 + bk + 0) * N_O + n];
      b1[1] = wlds[(4 + bk + 1) * N_O + n];

      v8f c = {};
      c = __builtin_amdgcn_wmma_f32_16x16x4_f32(false, a0, false, b0,
                                                (short)0, c, false, false);
      c = __builtin_amdgcn_wmma_f32_16x16x4_f32(false, a1, false, b1,
                                                (short)0, c, false, false);
      // C/D layout: VGPR v -> row M = v + 8*(lane>=16), col = lane&15
      #pragma unroll
      for (int v = 0; v < 8; ++v) {
        const int row = mt * 16 + v + 8 * (lane >> 4);
        votes[row * VROW + nt * 16 + bn] = c[v];
      }
    }
    __syncthreads();   // votes for this i complete in LDS

    // ---- routing math for this i
    float lg[8];
    float* lrow = logits + ((size_t)bb * N_IN_ + i) * N_OUT_;
    #pragma unroll
    for (int jj = 0; jj < 8; ++jj) {
      const int j = jq * 8 + jj;
      float l = lrow[j];
      if (r > 0) {   // deferred logits update: agreement with previous activation
        const float* av = act_in + ((size_t)bb * N_OUT_ + j) * OUT_A_;
        const float* vv = votes + bb * VROW + j * 16;
        float d = 0.0f;
        #pragma unroll
        for (int a = 0; a < 16; ++a) d += vv[a] * av[a];
        l += d;
        lrow[j] = l;
      }
      lg[jj] = l;
    }
    // leaky softmax over 32 j's (4 cooperating lanes) + leak logit 0
    float m = lg[0];
    #pragma unroll
    for (int jj = 1; jj < 8; ++jj) m = fmaxf(m, lg[jj]);
    m = fmaxf(m, __shfl_xor(m, 1, 32));
    m = fmaxf(m, __shfl_xor(m, 2, 32));
    m = fmaxf(m, 0.0f);
    float e[8], s = 0.0f;
    #pragma unroll
    for (int jj = 0; jj < 8; ++jj) { e[jj] = __expf(lg[jj] - m); s += e[jj]; }
    s += __shfl_xor(s, 1, 32);
    s += __shfl_xor(s, 2, 32);
    s += __expf(-m);                       // leak term
    const float inv = 1.0f / s;

    #pragma unroll
    for (int jj = 0; jj < 8; ++jj) {
      const int j = jq * 8 + jj;
      const float route = e[jj] * inv;
      const float* vv = votes + bb * VROW + j * 16;
      #pragma unroll
      for (int a = 0; a < 16; ++a) acc[jj][a] += route * vv[a];
    }
    __syncthreads();   // protect votes/wlds before next i overwrites
  }

  // ---- store this block's preact partial (deterministic, no atomics)
  float* pslot = partial + (size_t)blockIdx.x * PRE_N;
  #pragma unroll
  for (int jj = 0; jj < 8; ++jj) {
    const int j = jq * 8 + jj;
    float* p = pslot + ((size_t)bb * N_OUT_ + j) * OUT_A_;
    #pragma unroll
    for (int a = 0; a < 16; ++a) p[a] = acc[jj][a];
  }
}

// -------------------------------------------- reduce partials + bias + squash
__global__ __launch_bounds__(256) void k_finish(
    const float* __restrict__ partial, const float* __restrict__ bias,
    float* __restrict__ act_out)
{
  const int idx = blockIdx.x * 256 + threadIdx.x;   // 0 .. 32767 = b*512+j*16+a
  float v = 0.0f;
  for (int s = 0; s < NPART; ++s) v += partial[(size_t)s * PRE_N + idx];
  v += bias[idx & (N_O - 1)];          // bias[j*16+a]
  float sq = v * v;                    // ||v||^2 over 16 atoms (lane groups)
  sq += __shfl_xor(sq, 1, 32);
  sq += __shfl_xor(sq, 2, 32);
  sq += __shfl_xor(sq, 4, 32);
  sq += __shfl_xor(sq, 8, 32);
  const float norm = sqrtf(sq);
  // v/||v|| * ||v||^2/(1+||v||^2) == v * ||v|| / (1+||v||^2)
  act_out[idx] = v * (norm / (1.0f + sq));
}

// ---------------------------------------------------------------- launch
extern "C" void kernel_launch(void* const* d_in, const int* in_sizes, int n_in,
                              void* d_out, int out_size, void* d_ws, size_t ws_size,
                              hipStream_t stream) {
  (void)in_sizes; (void)n_in; (void)out_size; (void)ws_size;
  const float* x    = (const float*)d_in[0];   // [64, 2048, 8]
  const float* W    = (const float*)d_in[1];   // [2048, 8, 512]
  const float* bias = (const float*)d_in[2];   // [32, 16]

  float* logits  = (float*)d_ws;                                  // 4,194,304 f32
  float* partial = logits + (size_t)B_DIM * N_IN_ * N_OUT_;       // 4,194,304 f32
  float* act     = partial + (size_t)NPART * PRE_N;               //    32,768 f32

  k_init<<<1024, 256, 0, stream>>>(logits, B_DIM * N_IN_ * N_OUT_);

  for (int r = 0; r < NROUTE; ++r) {
    k_route<<<N_IN_ / I_PER_WG, 256, 0, stream>>>(
        x, W, logits, (r == 0) ? nullptr : act, partial, r);
    float* out = (r == NROUTE - 1) ? (float*)d_out : act;
    k_finish<<<PRE_N / 256, 256, 0, stream>>>(partial, bias, out);
  }
}